// RGCN_86577950753138
// MI455X (gfx1250) — compile-verified
//
#include <hip/hip_runtime.h>

// RGCN on MI455X (gfx1250, wave32).
// Aggregate-first rewrite:  segment_sum(x[src]) @ W  (halves matmul FLOPs vs edge-space,
// scatter runs on raw 128-dim features; x and A buffers are 51 MB each -> L2-resident,
// so gathers and f32 atomics resolve in the 192 MB L2, not HBM).
// GEMMs stay fp32 via V_WMMA_F32_16X16X4_F32 (
// system is atomic/L2 bound so the f32 WMMA rate is not the bottleneck).

#define Nn   100000
#define Rr   8
#define Ee   200000
#define Dk   128      // K dim of every GEMM (DIN = DH = 128)

typedef __attribute__((ext_vector_type(2))) float v2f;
typedef __attribute__((ext_vector_type(8))) float v8f;

// ---------------- utility kernels ----------------

__global__ void zero_kernel(float* __restrict__ p, int n) {
    int i = blockIdx.x * blockDim.x + threadIdx.x;
    int stride = gridDim.x * blockDim.x;
    for (; i < n; i += stride) p[i] = 0.0f;
}

// deg[r*N + dst[r*E+e]] += 1
__global__ void degree_kernel(const int* __restrict__ dst, float* __restrict__ deg) {
    int tid = blockIdx.x * blockDim.x + threadIdx.x;
    if (tid >= Rr * Ee) return;
    int r = tid / Ee;
    atomicAdd(&deg[r * Nn + dst[tid]], 1.0f);
}

__global__ void invert_kernel(float* __restrict__ deg, int n) {
    int i = blockIdx.x * blockDim.x + threadIdx.x;
    if (i >= n) return;
    float v = deg[i];
    deg[i] = (v > 0.0f) ? (1.0f / v) : 0.0f;
}

// One wave per edge; lane covers 4 consecutive floats (float4 gather = global_load_b128,
// 4 global_atomic_add_f32 resolved in L2).
__global__ void scatter128_kernel(const float* __restrict__ X,
                                  const int* __restrict__ src,
                                  const int* __restrict__ dst,
                                  float* __restrict__ A) {
    int tid  = blockIdx.x * blockDim.x + threadIdx.x;
    int edge = tid >> 5;
    int lane = tid & 31;
    if (edge >= Ee) return;
    int s = src[edge];
    int d = dst[edge];
    const float4 v = *(const float4*)(X + s * Dk + lane * 4);
    float* ap = A + d * Dk + lane * 4;
    atomicAdd(ap + 0, v.x);
    atomicAdd(ap + 1, v.y);
    atomicAdd(ap + 2, v.z);
    atomicAdd(ap + 3, v.w);
}

__global__ void bias_act_kernel(float* __restrict__ H, const float* __restrict__ b,
                                int D, int total, int relu) {
    int i = blockIdx.x * blockDim.x + threadIdx.x;
    int stride = gridDim.x * blockDim.x;
    for (; i < total; i += stride) {
        float v = H[i] + b[i & (D - 1)];   // D is 128 or 64 (pow2)
        H[i] = relu ? fmaxf(v, 0.0f) : v;
    }
}

// ---------------- WMMA GEMM: C[N,DOUT] += (A[N,128] * scale[N]?) @ W[128,DOUT] ----------
// One block = STRIPS consecutive 16-row strips (B fragments preloaded once into 64 VGPRs
// per wave and reused across all strips). One wave per 16-column tile:
// blockDim = 32*(DOUT/16). grid.x = (N/16)/STRIPS.
// Fragment layouts per ISA 7.12.2 (f32):
//   A 16x4 : vgpr0 = K{0|2} across lane halves, vgpr1 = K{1|3}; row = lane%16
//   B 4x16 : same K split; col = lane%16
//   C 16x16: vgpr g -> row (g | g+8) per lane half; col = lane%16

#define LSTR 132   // LDS row stride (floats): 16B-aligned float4 fills, conflict-free col reads

template<int DOUT, int STRIPS>
__global__ void __launch_bounds__(256)
gemm_acc_kernel(const float* __restrict__ A, const float* __restrict__ W,
                const float* __restrict__ scale, float* __restrict__ C, int useScale) {
    __shared__ float As[STRIPS * 16 * LSTR];

    const int tid  = threadIdx.x;
    const int wave = tid >> 5;
    const int lane = tid & 31;
    const int half = lane >> 4;      // 0: lanes 0-15, 1: lanes 16-31
    const int l16  = lane & 15;
    const int kk   = half * 2;       // per-half K offset inside a 4-wide step
    const int col0 = wave * 16;
    const int rowBase = blockIdx.x * (16 * STRIPS);

    // Stage all STRIPS A strips into LDS (float4 loads), applying inv_deg row scaling.
    const int nvec = STRIPS * 16 * (Dk / 4);
    for (int idx = tid; idx < nvec; idx += blockDim.x) {
        int m  = idx >> 5;                 // row within block: 0..16*STRIPS-1
        int k4 = (idx & 31) * 4;
        int grow = rowBase + m;
        float4 v = *(const float4*)(A + grow * Dk + k4);
        float sc = useScale ? scale[grow] : 1.0f;
        float* p = &As[m * LSTR + k4];
        p[0] = v.x * sc; p[1] = v.y * sc; p[2] = v.z * sc; p[3] = v.w * sc;
    }

    // Preload all 32 B fragments; DOUT is a compile-time constant so these are
    // base + immediate-offset scalar loads (no 64-bit address math).
    v2f bfrag[32];
#pragma unroll
    for (int ks = 0; ks < 32; ++ks) {
        int k0 = ks * 4 + kk;
        bfrag[ks].x = W[(k0 + 0) * DOUT + col0 + l16];
        bfrag[ks].y = W[(k0 + 1) * DOUT + col0 + l16];
    }

    __syncthreads();

#pragma unroll
    for (int s = 0; s < STRIPS; ++s) {
        const int row0 = rowBase + s * 16;

        // Load the C accumulator tile (layer-level accumulation across relations).
        v8f c;
#pragma unroll
        for (int g = 0; g < 8; ++g)
            c[g] = C[(row0 + g + half * 8) * DOUT + col0 + l16];

        // 32 chained V_WMMA_F32_16X16X4_F32 steps cover K = 128.
#pragma unroll
        for (int ks = 0; ks < 32; ++ks) {
            v2f a;
            const int base = (s * 16 + l16) * LSTR + ks * 4 + kk;
            a.x = As[base];
            a.y = As[base + 1];
            c = __builtin_amdgcn_wmma_f32_16x16x4_f32(
                    /*neg_a=*/false, a, /*neg_b=*/false, bfrag[ks],
                    /*c_mod=*/(short)0, c, /*reuse_a=*/false, /*reuse_b=*/false);
        }

#pragma unroll
        for (int g = 0; g < 8; ++g)
            C[(row0 + g + half * 8) * DOUT + col0 + l16] = c[g];
    }
}

// ---------------- host-side orchestration ----------------

extern "C" void kernel_launch(void* const* d_in, const int* in_sizes, int n_in,
                              void* d_out, int out_size, void* d_ws, size_t ws_size,
                              hipStream_t stream) {
    const float* x     = (const float*)d_in[0];
    const int*   src   = (const int*)  d_in[1];
    const int*   dst   = (const int*)  d_in[2];
    const float* W1    = (const float*)d_in[3];
    const float* loop1 = (const float*)d_in[4];
    const float* b1    = (const float*)d_in[5];
    const float* W2    = (const float*)d_in[6];
    const float* loop2 = (const float*)d_in[7];
    const float* b2    = (const float*)d_in[8];
    const float* W3    = (const float*)d_in[9];
    const float* loop3 = (const float*)d_in[10];
    const float* b3    = (const float*)d_in[11];
    float* out = (float*)d_out;

    // Workspace layout: inv_deg[R*N] | A[N*128] | H1[N*128] | H2[N*128]  (~157 MB)
    float* inv = (float*)d_ws;
    float* A   = inv + Rr * Nn;
    float* H1  = A   + Nn * Dk;
    float* H2  = H1  + Nn * Dk;

    auto zero = [&](float* p, int n) {
        zero_kernel<<<2048, 256, 0, stream>>>(p, n);
    };

    constexpr int STRIPS = 5;                 // 6250 strips = 1250 blocks * 5
    constexpr int GBLK   = (Nn / 16) / STRIPS;

    // ---- per-relation inverse in-degrees (degree-0 -> 0, matching DGL 'right' norm) ----
    zero(inv, Rr * Nn);
    degree_kernel<<<(Rr * Ee + 255) / 256, 256, 0, stream>>>(dst, inv);
    invert_kernel<<<(Rr * Nn + 255) / 256, 256, 0, stream>>>(inv, Rr * Nn);

    // ---------------- layer 1 & 2 (DOUT = 128) ----------------
    auto layer128 = [&](const float* X, const float* W, const float* loopW,
                        const float* bias, float* H) {
        zero(H, Nn * 128);
        for (int r = 0; r < Rr; ++r) {
            zero(A, Nn * Dk);
            scatter128_kernel<<<Ee / 8, 256, 0, stream>>>(X, src + r * Ee, dst + r * Ee, A);
            gemm_acc_kernel<128, STRIPS><<<GBLK, 256, 0, stream>>>(
                A, W + r * Dk * 128, inv + r * Nn, H, 1);
        }
        gemm_acc_kernel<128, STRIPS><<<GBLK, 256, 0, stream>>>(X, loopW, nullptr, H, 0);
        bias_act_kernel<<<2048, 256, 0, stream>>>(H, bias, 128, Nn * 128, 1);
    };

    layer128(x,  W1, loop1, b1, H1);
    layer128(H1, W2, loop2, b2, H2);

    // ---------------- layer 3 (DOUT = 64, no ReLU, output -> d_out) ----------------
    zero(out, Nn * 64);
    for (int r = 0; r < Rr; ++r) {
        zero(A, Nn * Dk);
        scatter128_kernel<<<Ee / 8, 256, 0, stream>>>(H2, src + r * Ee, dst + r * Ee, A);
        gemm_acc_kernel<64, STRIPS><<<GBLK, 128, 0, stream>>>(
            A, W3 + r * Dk * 64, inv + r * Nn, out, 1);
    }
    gemm_acc_kernel<64, STRIPS><<<GBLK, 128, 0, stream>>>(H2, loop3, nullptr, out, 0);
    bias_act_kernel<<<2048, 256, 0, stream>>>(out, b3, 64, Nn * 64, 0);
}